// SetCriterion_50130858279211
// MI455X (gfx1250) — compile-verified
//
#include <hip/hip_runtime.h>
#include <math.h>

namespace {
constexpr int kB = 16384;
constexpr int kF = 512;
constexpr int kT = 256;
constexpr int kN = 256;
}

__device__ __forceinline__ float waveAdd(float v) {
#pragma unroll
  for (int off = 16; off >= 1; off >>= 1) v += __shfl_xor(v, off, 32);
  return v;
}

__device__ __forceinline__ float waveMul(float v) {
#pragma unroll
  for (int off = 16; off >= 1; off >>= 1) v *= __shfl_xor(v, off, 32);
  return v;
}

// Per-block criterion. ATOMIC=false: write double4 partial per block (deterministic
// path). ATOMIC=true: accumulate into 4 global doubles (fallback if d_ws is tiny).
template <bool ATOMIC>
__global__ __launch_bounds__(256) void sc_main(
    const float* __restrict__ pred_class,
    const float* __restrict__ pred_v,
    const float* __restrict__ targets,
    double* __restrict__ outp) {
  __shared__ float s_v[kN];       // pred_v row (async-copied)
  __shared__ float s_pick[8];     // per-wave picked-logprob sums
  __shared__ float s_prod[8];     // per-wave products
  __shared__ float s_lv[8];       // per-wave lossv1 sums
  __shared__ unsigned s_cnt[8];   // per-wave mask popcounts

  const int b = blockIdx.x;
  const int t = threadIdx.x;
  const int lane = t & 31;
  const int wave = t >> 5;

  // ---- CDNA5 async global->LDS copy of the pred_v row (ASYNCcnt path). ----
  // LDS byte address in a VGPR; global address = 64-bit SGPR base + 32-bit VGPR offset
  {
    unsigned lds_off = (unsigned)(unsigned long long)&s_v[t];
    unsigned long long rowp = (unsigned long long)(pred_v + (size_t)b * kN);
    unsigned voff = (unsigned)(t * 4);
    asm volatile("global_load_async_to_lds_b32 %0, %1, %2"
                 :: "v"(lds_off), "v"(voff), "s"(rowp)
                 : "memory");
  }

  // ---- Overlap with the async copy: 2-class log-softmax ----
  // pred_class row layout: (f, c) at f*2 + c; thread t handles f=t and f=t+256.
  const float2* cls = (const float2*)(pred_class + (size_t)b * (kF * 2));
  float2 l0 = cls[t];        // f = t      < 256  -> true label 1
  float2 l1 = cls[t + kT];   // f = t + 256       -> true label 0

  float m0 = fmaxf(l0.x, l0.y);
  float picked0 = l0.y - (m0 + logf(expf(l0.x - m0) + expf(l0.y - m0)));
  float m1 = fmaxf(l1.x, l1.y);
  float picked1 = l1.x - (m1 + logf(expf(l1.x - m1) + expf(l1.y - m1)));
  float wsum = waveAdd(picked0 + picked1);
  if (lane == 0) s_pick[wave] = wsum;

  // mask = (argmax == 1); jnp.argmax picks index 0 on ties, so strict >
  bool mask = l0.y > l0.x;
#if __has_builtin(__builtin_amdgcn_ballot_w32)
  unsigned bal = __builtin_amdgcn_ballot_w32(mask);
#else
  unsigned bal = (unsigned)__ballot(mask);
#endif
  unsigned prefix = __popc(bal & ((1u << lane) - 1u));  // masked lanes before me
  if (lane == 0) s_cnt[wave] = __popc(bal);

  float tgt = targets[(size_t)b * kT + t];

  // ---- wait for our async LDS writes, then make row visible block-wide ----
#if __has_builtin(__builtin_amdgcn_s_wait_asynccnt)
  __builtin_amdgcn_s_wait_asynccnt(0);
#else
  asm volatile("s_wait_asynccnt 0" ::: "memory");
#endif
  __syncthreads();

  // cross-wave scan offsets + total mask count
  unsigned wave_off = 0, t_count = 0;
#pragma unroll
  for (int w = 0; w < 8; ++w) {
    unsigned c = s_cnt[w];
    wave_off += (w < wave) ? c : 0u;
    t_count += c;
  }

  // matched gather: idx = inclusive_cumsum - 1 = wave_off + prefix (mask lanes)
  float contrib = 0.f;
  if (mask) {
    float d = s_v[wave_off + prefix] - tgt;
    contrib = d * d;
  }
  float lsum = waveAdd(contrib);
  float wprod = waveMul(s_v[t]);
  if (lane == 0) { s_lv[wave] = lsum; s_prod[wave] = wprod; }
  __syncthreads();

  if (t == 0) {
    float pick_s = 0.f, lossv1 = 0.f, prod = 1.f;
#pragma unroll
    for (int w = 0; w < 8; ++w) {
      pick_s += s_pick[w];
      lossv1 += s_lv[w];
      prod *= s_prod[w];
    }
    float e = prod - 1.0f;
    bool has = t_count > 0;
    double p0 = (double)pick_s;                                   // sum picked
    double p1 = (double)(e * e);                                  // (prod-1)^2
    double p2 = has ? (double)(lossv1 * (1.0f / kT)) : 0.0;       // lossv1/T
    double p3 = has ? (double)t_count * (1.0 / kT) : 0.0;         // t_count/T
    if (ATOMIC) {
      atomicAdd(&outp[0], p0);
      atomicAdd(&outp[1], p1);
      atomicAdd(&outp[2], p2);
      atomicAdd(&outp[3], p3);
    } else {
      double4* part = (double4*)outp;
      part[b] = make_double4(p0, p1, p2, p3);
    }
  }
}

__device__ __forceinline__ void sc_finalize(double a0, double a1, double a2,
                                            double a3, float* out) {
  const double Bd = (double)kB;
  double loss_label = -a0 / (Bd * (double)kF);
  double loss_multi = a1 / Bd;
  double loss_true  = a2 / Bd;
  double f1         = a3 / Bd;
  double r = loss_multi + loss_true;
  out[0] = (float)(loss_label + r);
  out[1] = (float)loss_label;
  out[2] = (float)r;
  out[3] = (float)loss_multi;
  out[4] = (float)loss_true;
  out[5] = (float)f1;
}

// Deterministic stage 1: 64 blocks x 256 threads, fixed-pairing LDS tree.
__global__ __launch_bounds__(256) void sc_reduce1(const double4* __restrict__ part,
                                                  double4* __restrict__ mid) {
  __shared__ double4 sm[256];
  const int t = threadIdx.x;
  sm[t] = part[(size_t)blockIdx.x * 256 + t];
  __syncthreads();
#pragma unroll
  for (int s = 128; s > 0; s >>= 1) {
    if (t < s) {
      double4 a = sm[t], c = sm[t + s];
      a.x += c.x; a.y += c.y; a.z += c.z; a.w += c.w;
      sm[t] = a;
    }
    __syncthreads();
  }
  if (t == 0) mid[blockIdx.x] = sm[0];
}

// Deterministic stage 2: 1 block x 64 threads, tree reduce + finalize.
__global__ __launch_bounds__(64) void sc_reduce2(const double4* __restrict__ mid,
                                                 float* __restrict__ out) {
  __shared__ double4 sm[64];
  const int t = threadIdx.x;
  sm[t] = mid[t];
  __syncthreads();
#pragma unroll
  for (int s = 32; s > 0; s >>= 1) {
    if (t < s) {
      double4 a = sm[t], c = sm[t + s];
      a.x += c.x; a.y += c.y; a.z += c.z; a.w += c.w;
      sm[t] = a;
    }
    __syncthreads();
  }
  if (t == 0) sc_finalize(sm[0].x, sm[0].y, sm[0].z, sm[0].w, out);
}

// Atomic-path helpers (fallback when d_ws is too small for partials).
__global__ void sc_zero(double* acc) {
  if (threadIdx.x < 8) acc[threadIdx.x] = 0.0;
}

__global__ void sc_final(const double* __restrict__ acc, float* __restrict__ out) {
  if (threadIdx.x == 0 && blockIdx.x == 0)
    sc_finalize(acc[0], acc[1], acc[2], acc[3], out);
}

extern "C" void kernel_launch(void* const* d_in, const int* in_sizes, int n_in,
                              void* d_out, int out_size, void* d_ws, size_t ws_size,
                              hipStream_t stream) {
  (void)in_sizes; (void)n_in; (void)out_size;
  const float* pred_class = (const float*)d_in[0];
  const float* pred_v     = (const float*)d_in[1];
  const float* targets    = (const float*)d_in[2];
  float* out = (float*)d_out;

  const size_t need = (size_t)(kB + 64) * sizeof(double4);
  if (ws_size >= need) {
    // Deterministic tree-reduction path.
    double4* part = (double4*)d_ws;      // kB entries
    double4* mid  = part + kB;           // 64 entries
    sc_main<false><<<dim3(kB), dim3(256), 0, stream>>>(
        pred_class, pred_v, targets, (double*)part);
    sc_reduce1<<<dim3(64), dim3(256), 0, stream>>>(part, mid);
    sc_reduce2<<<dim3(1), dim3(64), 0, stream>>>(mid, out);
  } else {
    // Fallback: f64 atomic accumulation (needs only 64 B of scratch).
    double* acc = (double*)d_ws;
    sc_zero<<<dim3(1), dim3(32), 0, stream>>>(acc);
    sc_main<true><<<dim3(kB), dim3(256), 0, stream>>>(
        pred_class, pred_v, targets, acc);
    sc_final<<<dim3(1), dim3(1), 0, stream>>>(acc, out);
  }
}